// Binarize_82068235092640
// MI455X (gfx1250) — compile-verified
//
#include <hip/hip_runtime.h>
#include <stdint.h>

// Binarize + packbits (big-endian bit order), CDNA5 / gfx1250.
// x: [4096, 8192] f32, depth: [3] f32  ->  out: [4096, 3, 1024] uint8.
// Wave32 ballot does the bit packing; async global->LDS copies stream the input.

#define WAVES_PER_BLOCK 8
#define THREADS_PER_BLOCK (WAVES_PER_BLOCK * 32)
#define TILE_ELEMS 1024            // floats per tile -> 128 out bytes per threshold
#define TILES_PER_WAVE 4
#define TILES_PER_ROW 8            // 8192 cols / 1024

#if defined(__gfx1250__) && __has_builtin(__builtin_amdgcn_global_load_async_to_lds_b128)
#define USE_ASYNC_LDS 1
#endif

#if defined(USE_ASYNC_LDS)
#if __has_builtin(__builtin_amdgcn_s_wait_asynccnt)
#define WAIT_ASYNC(n) __builtin_amdgcn_s_wait_asynccnt(n)
#else
#define WAIT_ASYNC(n) asm volatile("s_wait_asynccnt " #n ::: "memory")
#endif

typedef int v4i __attribute__((ext_vector_type(4)));
typedef __attribute__((address_space(1))) v4i glb_v4i;   // global int4
typedef __attribute__((address_space(3))) v4i lds_v4i;   // LDS int4
typedef __attribute__((address_space(1))) char as1_char;
typedef __attribute__((address_space(3))) char as3_char;

// Copy one 4 KiB tile global->LDS: 8 x (32 lanes x 16B) async B128 transfers.
__device__ __forceinline__ void issue_tile_async(const float* g_tile, float* l_tile, int lane) {
  as1_char* g = (as1_char*)g_tile + lane * 16;
  as3_char* l = (as3_char*)l_tile + lane * 16;
#pragma unroll
  for (int k = 0; k < 8; ++k) {
    __builtin_amdgcn_global_load_async_to_lds_b128((glb_v4i*)(g + k * 512),
                                                   (lds_v4i*)(l + k * 512),
                                                   0, 0);
  }
}
#endif  // USE_ASYNC_LDS

__device__ __forceinline__ unsigned ballot32(int pred) {
#if __has_builtin(__builtin_amdgcn_ballot_w32)
  return __builtin_amdgcn_ballot_w32(pred);
#else
  return (unsigned)__ballot(pred);
#endif
}

__global__ __launch_bounds__(THREADS_PER_BLOCK)
void binpack_kernel(const float* __restrict__ x, const float* __restrict__ depth,
                    uint32_t* __restrict__ out, int total_tiles) {
#if defined(USE_ASYNC_LDS)
  __shared__ float lds[WAVES_PER_BLOCK][2][TILE_ELEMS];   // 64 KiB: double buffer per wave
#endif
  const int lane = threadIdx.x & 31;
  const int w    = threadIdx.x >> 5;
  const int tile0 = (blockIdx.x * WAVES_PER_BLOCK + w) * TILES_PER_WAVE;
  if (tile0 + TILES_PER_WAVE > total_tiles) return;

  const float th0 = depth[0];
  const float th1 = depth[1];
  const float th2 = depth[2];

#if defined(USE_ASYNC_LDS)
  issue_tile_async(x + (size_t)tile0 * TILE_ELEMS, &lds[w][0][0], lane);
#endif

#pragma unroll
  for (int t = 0; t < TILES_PER_WAVE; ++t) {
    const int tile = tile0 + t;
    const float* src;
#if defined(USE_ASYNC_LDS)
    if (t + 1 < TILES_PER_WAVE) {
      issue_tile_async(x + (size_t)(tile + 1) * TILE_ELEMS, &lds[w][(t + 1) & 1][0], lane);
      WAIT_ASYNC(8);   // the 8 newest (next tile) may be pending; current tile has landed
    } else {
      WAIT_ASYNC(0);
    }
    src = &lds[w][t & 1][0];
#else
    src = x + (size_t)tile * TILE_ELEMS;
    if (t + 1 < TILES_PER_WAVE)
      __builtin_prefetch(x + (size_t)(tile + 1) * TILE_ELEMS + lane * 32, 0, 0);
#endif

    // 32 chunks of 32 elements; ballot packs each chunk into one uniform dword.
    // Select chunk c's raw mask into lane c (v_cndmask), then per-lane
    // bswap(bitrev32(m)) converts ballot bit order to numpy big-endian packbits.
    unsigned r0 = 0, r1 = 0, r2 = 0;
#pragma unroll
    for (int c = 0; c < 32; ++c) {
      const float v = src[c * 32 + lane];
      const unsigned m0 = ballot32(v > th0);
      const unsigned m1 = ballot32(v > th1);
      const unsigned m2 = ballot32(v > th2);
      const bool mine = (lane == c);
      r0 = mine ? m0 : r0;
      r1 = mine ? m1 : r1;
      r2 = mine ? m2 : r2;
    }

    const int row = tile / TILES_PER_ROW;
    const int cc  = tile % TILES_PER_ROW;
    uint32_t* o = out + (size_t)row * (3 * 256) + cc * 32 + lane;
    o[0]   = __builtin_bswap32(__builtin_bitreverse32(r0));   // d = 0
    o[256] = __builtin_bswap32(__builtin_bitreverse32(r1));   // d = 1
    o[512] = __builtin_bswap32(__builtin_bitreverse32(r2));   // d = 2
  }
}

extern "C" void kernel_launch(void* const* d_in, const int* in_sizes, int n_in,
                              void* d_out, int out_size, void* d_ws, size_t ws_size,
                              hipStream_t stream) {
  (void)n_in; (void)d_ws; (void)ws_size; (void)out_size;
  const float* x     = (const float*)d_in[0];
  const float* depth = (const float*)d_in[1];
  uint32_t*    out   = (uint32_t*)d_out;

  const int n_elems     = in_sizes[0];                 // 4096 * 8192
  const int total_tiles = n_elems / TILE_ELEMS;        // 32768
  const int tiles_per_block = WAVES_PER_BLOCK * TILES_PER_WAVE;
  const int blocks = (total_tiles + tiles_per_block - 1) / tiles_per_block;  // 1024

  binpack_kernel<<<dim3(blocks), dim3(THREADS_PER_BLOCK), 0, stream>>>(
      x, depth, out, total_tiles);
}